// PersonalityGNN_45028437131739
// MI455X (gfx1250) — compile-verified
//
#include <hip/hip_runtime.h>
#include <math.h>

// ---------------------------------------------------------------------------
// PersonalityGNN forward for MI455X (gfx1250, wave32, WMMA).
// Dense layers use V_WMMA_F32_16X16X4_F32 (exact fp32 = reference precision).
// BatchNorm is folded to per-column scale/shift at stats time so the GEMM
// prologue is a single FMA (keeps the WMMA loop free of rsq/cndmask chains).
// Graph ops (segment sum / segment softmax) use float atomics.
// Workspace: ~350 MB laid out manually with phase reuse.
// ---------------------------------------------------------------------------

#define NU_   50000
#define NPR_  50000
#define NN_   100000          // NU_ + NPR_
#define EDG_  600000
#define ELP_  (EDG_ + NN_)    // edges + self loops (GATv2)
#define HID_  128

typedef float v2f __attribute__((ext_vector_type(2)));
typedef float v8f __attribute__((ext_vector_type(8)));

__device__ __forceinline__ float gelu_f(float x) {
  // exact GELU: x * 0.5 * (1 + erf(x / sqrt(2)))
  return 0.5f * x * (1.0f + erff(x * 0.70710678118654752f));
}
// float <-> order-preserving unsigned (for atomicMax on floats)
__device__ __forceinline__ unsigned f2o(float f) {
  unsigned u = __float_as_uint(f);
  return (u & 0x80000000u) ? ~u : (u | 0x80000000u);
}
__device__ __forceinline__ float o2f(unsigned u) {
  return (u & 0x80000000u) ? __uint_as_float(u & 0x7fffffffu)
                           : __uint_as_float(~u);
}

// -------------------------------------------------------------------- stats
// One block (256 threads) per column: population mean/var over rows, folded
// into affine BN form:  scale = g * rsqrt(var+1e-5);  shift = b - mean*scale.
__global__ void colstats_kernel(const float* __restrict__ X, int rows, int cols,
                                const float* __restrict__ g, const float* __restrict__ b,
                                float* __restrict__ scale, float* __restrict__ shift) {
  int c = blockIdx.x;
  float s = 0.f, s2 = 0.f;
  for (int r = threadIdx.x; r < rows; r += 256) {
    float v = X[(size_t)r * cols + c];
    s += v; s2 += v * v;
  }
  __shared__ float sh[256], sh2[256];
  sh[threadIdx.x] = s; sh2[threadIdx.x] = s2;
  __syncthreads();
  for (int o = 128; o > 0; o >>= 1) {
    if ((int)threadIdx.x < o) { sh[threadIdx.x] += sh[threadIdx.x + o];
                                sh2[threadIdx.x] += sh2[threadIdx.x + o]; }
    __syncthreads();
  }
  if (threadIdx.x == 0) {
    float m  = sh[0] / (float)rows;
    float va = sh2[0] / (float)rows - m * m;
    float sc = g[c] * rsqrtf(va + 1e-5f);
    scale[c] = sc;
    shift[c] = b[c] - m * sc;
  }
}

// --------------------------------------------------------------------- GEMM
enum { PRE_NONE = 0, PRE_BN = 1, PRE_GELU = 2, PRE_BN_GELU = 3, PRE_ROWSCALE = 4 };

// Y[rows x OD] = pre(X[rows x KD]) @ W[OD x KD]^T + bias (+ res_scale*res)
// 4 waves / block; each wave -> one 16x16 tile via V_WMMA_F32_16X16X4_F32.
// A frag: lane&15 = M, lane>>4 selects K pair {k,k+1}/{k+2,k+3} in v[0],v[1].
// B frag: same layout on W rows (lane&15 = N) -> computes X @ W^T.
template <int KD, int PRE>
__global__ void __launch_bounds__(128)
gemm_kernel(const float* __restrict__ X, const float* __restrict__ W,
            const float* __restrict__ bias, float* __restrict__ Y, int OD,
            const float* __restrict__ bnS, const float* __restrict__ bnB,
            const float* __restrict__ rowscale,
            const float* __restrict__ res, float res_scale) {
  int wave = threadIdx.x >> 5;
  int lane = threadIdx.x & 31;
  int l15  = lane & 15;
  int half = lane >> 4;
  int row0 = blockIdx.x * 16;
  int n0   = blockIdx.y * 64 + wave * 16;

  int arow = row0 + l15;
  float rsc = 1.0f;
  if (PRE == PRE_ROWSCALE) rsc = rowscale[arow];

  v8f acc;
#pragma unroll
  for (int i = 0; i < 8; ++i) acc[i] = 0.0f;

  const float* xrow = X + (size_t)arow * KD;
  const float* wrow = W + (size_t)(n0 + l15) * KD;

#pragma unroll 4
  for (int k = 0; k < KD; k += 4) {
    int kk = k + half * 2;                        // even -> 8B aligned
    v2f A = *(const v2f*)(xrow + kk);
    v2f B = *(const v2f*)(wrow + kk);
    if (PRE == PRE_BN || PRE == PRE_BN_GELU) {
      v2f sc = *(const v2f*)(bnS + kk);
      v2f sb = *(const v2f*)(bnB + kk);
      A[0] = fmaf(A[0], sc[0], sb[0]);
      A[1] = fmaf(A[1], sc[1], sb[1]);
    }
    if (PRE == PRE_GELU || PRE == PRE_BN_GELU) { A[0] = gelu_f(A[0]); A[1] = gelu_f(A[1]); }
    if (PRE == PRE_ROWSCALE) { A[0] *= rsc; A[1] *= rsc; }
    acc = __builtin_amdgcn_wmma_f32_16x16x4_f32(false, A, false, B,
                                                (short)0, acc, false, false);
  }

  int col = n0 + l15;
  float bs = bias ? bias[col] : 0.0f;
  int rbase = row0 + half * 8;   // C/D: vgpr r -> M=r (lanes 0-15), M=r+8 (16-31)
#pragma unroll
  for (int r = 0; r < 8; ++r) {
    size_t idx = (size_t)(rbase + r) * OD + col;
    float v = acc[r] + bs;
    if (res) v += res_scale * res[idx];
    Y[idx] = v;
  }
}

// ------------------------------------------------------------- elementwise
// out = gelu( x*scale[c]+shift[c] (+ res) )
__global__ void bn_res_gelu_kernel(const float* __restrict__ T,
                                   const float* __restrict__ res,
                                   const float* __restrict__ sc, const float* __restrict__ sb,
                                   float* __restrict__ out, long total, int cols) {
  long i = (long)blockIdx.x * blockDim.x + threadIdx.x;
  if (i >= total) return;
  int c = (int)(i % cols);
  float x = fmaf(T[i], sc[c], sb[c]);
  if (res) x += res[i];
  out[i] = gelu_f(x);
}

__global__ void fill_f32(float* __restrict__ p, long n, float v) {
  long i = (long)blockIdx.x * blockDim.x + threadIdx.x;
  if (i < n) p[i] = v;
}
__global__ void fill_u32(unsigned* __restrict__ p, long n, unsigned v) {
  long i = (long)blockIdx.x * blockDim.x + threadIdx.x;
  if (i < n) p[i] = v;
}

// -------------------------------------------------------------- graph ops
__global__ void deg_kernel(const int* __restrict__ dst, int ne, float* __restrict__ deg) {
  int e = blockIdx.x * blockDim.x + threadIdx.x;
  if (e < ne) atomicAdd(&deg[dst[e]], 1.0f);
}
__global__ void recip_deg_kernel(float* __restrict__ deg, int n) {
  int i = blockIdx.x * blockDim.x + threadIdx.x;
  if (i < n) deg[i] = 1.0f / fmaxf(deg[i], 1.0f);
}
__global__ void scatter_agg_kernel(const float* __restrict__ x,
                                   const int* __restrict__ src, const int* __restrict__ dst,
                                   float* __restrict__ agg, long total) {
  long t = (long)blockIdx.x * blockDim.x + threadIdx.x;
  if (t >= total) return;
  int e = (int)(t >> 7), c = (int)(t & 127);
  atomicAdd(&agg[(size_t)dst[e] * HID_ + c], x[(size_t)src[e] * HID_ + c]);
}
// GATv2 scoring (edges + N self loops): score[e,h] = sum_c lrelu(xl[s]+xr[d])*att
__global__ void gat_score_kernel(const float* __restrict__ xl, const float* __restrict__ xr,
                                 const float* __restrict__ att,
                                 const int* __restrict__ src, const int* __restrict__ dst,
                                 int ne, int nloops,
                                 float* __restrict__ esc, unsigned* __restrict__ mU) {
  int t = blockIdx.x * blockDim.x + threadIdx.x;
  int tot = (ne + nloops) * 4;
  if (t >= tot) return;
  int e = t >> 2, h = t & 3;
  int s = (e < ne) ? src[e] : e - ne;
  int d = (e < ne) ? dst[e] : e - ne;
  const float* pl = xl + (size_t)s * HID_ + h * 32;
  const float* pr = xr + (size_t)d * HID_ + h * 32;
  const float* pa = att + h * 32;
  float sc = 0.f;
#pragma unroll 8
  for (int c = 0; c < 32; ++c) {
    float v = pl[c] + pr[c];
    v = (v >= 0.f) ? v : 0.2f * v;          // leaky_relu 0.2
    sc += v * pa[c];
  }
  esc[(size_t)e * 4 + h] = sc;
  atomicMax(&mU[(size_t)d * 4 + h], f2o(sc));
}
// TransformerConv scoring: (q[dst].k[src]) / sqrt(32)
__global__ void tc_score_kernel(const float* __restrict__ q, const float* __restrict__ k,
                                const int* __restrict__ src, const int* __restrict__ dst,
                                int ne, float* __restrict__ esc, unsigned* __restrict__ mU) {
  int t = blockIdx.x * blockDim.x + threadIdx.x;
  if (t >= ne * 4) return;
  int e = t >> 2, h = t & 3;
  int s = src[e], d = dst[e];
  const float* pq = q + (size_t)d * HID_ + h * 32;
  const float* pk = k + (size_t)s * HID_ + h * 32;
  float sc = 0.f;
#pragma unroll 8
  for (int c = 0; c < 32; ++c) sc += pq[c] * pk[c];
  sc *= 0.17677669529663687f;               // 1/sqrt(32)
  esc[(size_t)e * 4 + h] = sc;
  atomicMax(&mU[(size_t)d * 4 + h], f2o(sc));
}
__global__ void seg_exp_kernel(const int* __restrict__ dst, int ne, int nloops,
                               const unsigned* __restrict__ mU,
                               float* __restrict__ esc, float* __restrict__ z) {
  int t = blockIdx.x * blockDim.x + threadIdx.x;
  int tot = (ne + nloops) * 4;
  if (t >= tot) return;
  int e = t >> 2, h = t & 3;
  int d = (e < ne) ? dst[e] : e - ne;
  float ex = expf(esc[t] - o2f(mU[(size_t)d * 4 + h]));
  esc[t] = ex;
  atomicAdd(&z[(size_t)d * 4 + h], ex);
}
__global__ void seg_agg_kernel(const float* __restrict__ val,
                               const int* __restrict__ src, const int* __restrict__ dst,
                               int ne, int nloops,
                               const float* __restrict__ esc, const float* __restrict__ z,
                               float* __restrict__ out) {
  long t = (long)blockIdx.x * blockDim.x + threadIdx.x;
  long tot = (long)(ne + nloops) * HID_;
  if (t >= tot) return;
  int e = (int)(t >> 7), c = (int)(t & 127), h = c >> 5;
  int s = (e < ne) ? src[e] : e - ne;
  int d = (e < ne) ? dst[e] : e - ne;
  float alpha = esc[(size_t)e * 4 + h] / (z[(size_t)d * 4 + h] + 1e-16f);
  atomicAdd(&out[(size_t)d * HID_ + c], val[(size_t)s * HID_ + c] * alpha);
}
// x = gelu(LN(0.6*sage + 0.4*(aout [+addvec]))) + 0.8*x   (one block per row)
__global__ void combine_kernel(float* __restrict__ x, const float* __restrict__ xsage,
                               const float* __restrict__ aout,
                               const float* __restrict__ addvec,
                               const float* __restrict__ lng, const float* __restrict__ lnb) {
  int row = blockIdx.x, c = threadIdx.x;
  size_t idx = (size_t)row * HID_ + c;
  float iden = x[idx];
  float gp = aout[idx] + (addvec ? addvec[c] : 0.f);
  float v = 0.6f * xsage[idx] + 0.4f * gp;
  __shared__ float rs[HID_];
  __shared__ float mu_s, var_s;
  rs[c] = v; __syncthreads();
  for (int o = 64; o > 0; o >>= 1) { if (c < o) rs[c] += rs[c + o]; __syncthreads(); }
  if (c == 0) mu_s = rs[0] * (1.0f / HID_);
  __syncthreads();
  float dv = v - mu_s;
  rs[c] = dv * dv; __syncthreads();
  for (int o = 64; o > 0; o >>= 1) { if (c < o) rs[c] += rs[c + o]; __syncthreads(); }
  if (c == 0) var_s = rs[0] * (1.0f / HID_);
  __syncthreads();
  float y = lng[c] * dv * rsqrtf(var_s + 1e-5f) + lnb[c];
  x[idx] = gelu_f(y) + 0.8f * iden;
}
__global__ void concat_kernel(const float* __restrict__ x, const float* __restrict__ xu,
                              float* __restrict__ ue, int nu) {
  long t = (long)blockIdx.x * blockDim.x + threadIdx.x;
  long tot = (long)nu * 256;
  if (t >= tot) return;
  int row = (int)(t >> 8), c = (int)(t & 255);
  ue[t] = (c < 128) ? x[(size_t)row * 128 + c] : xu[(size_t)row * 128 + (c - 128)];
}
// final: fc3(5x128) on gelu(o1*sc+sb), LN over 5, sigmoid
__global__ void head_kernel(const float* __restrict__ o1,
                            const float* __restrict__ sc, const float* __restrict__ sb,
                            const float* __restrict__ W, const float* __restrict__ bias,
                            const float* __restrict__ og, const float* __restrict__ ob,
                            float* __restrict__ out, int rows) {
  int row = blockIdx.x * blockDim.x + threadIdx.x;
  if (row >= rows) return;
  float h[5];
#pragma unroll
  for (int j = 0; j < 5; ++j) h[j] = bias[j];
  const float* xr = o1 + (size_t)row * 128;
  for (int c = 0; c < 128; ++c) {
    float x = gelu_f(fmaf(xr[c], sc[c], sb[c]));
#pragma unroll
    for (int j = 0; j < 5; ++j) h[j] += W[j * 128 + c] * x;
  }
  float mu = 0.f;
#pragma unroll
  for (int j = 0; j < 5; ++j) mu += h[j];
  mu *= 0.2f;
  float va = 0.f;
#pragma unroll
  for (int j = 0; j < 5; ++j) { float d = h[j] - mu; va += d * d; }
  va *= 0.2f;
  float rstd = rsqrtf(va + 1e-5f);
#pragma unroll
  for (int j = 0; j < 5; ++j) {
    float y = og[j] * (h[j] - mu) * rstd + ob[j];
    out[(size_t)row * 5 + j] = 1.0f / (1.0f + expf(-y));
  }
}

// --------------------------------------------------------------- launchers
template <int KD, int PRE>
static void G(hipStream_t st, const float* X, int rows, const float* W, const float* b,
              float* Y, int OD,
              const float* bnS = nullptr, const float* bnB = nullptr,
              const float* rsc = nullptr, const float* res = nullptr, float rscale = 0.f) {
  dim3 grid(rows / 16, OD / 64);
  gemm_kernel<KD, PRE><<<grid, 128, 0, st>>>(X, W, b, Y, OD, bnS, bnB, rsc, res, rscale);
}
static void STATS(hipStream_t st, const float* X, int rows, int cols,
                  const float* g, const float* b, float* sc, float* sh) {
  colstats_kernel<<<cols, 256, 0, st>>>(X, rows, cols, g, b, sc, sh);
}
static void FILLF(hipStream_t st, float* p, long n, float v) {
  fill_f32<<<(int)((n + 255) / 256), 256, 0, st>>>(p, n, v);
}
static void FILLU(hipStream_t st, unsigned* p, long n, unsigned v) {
  fill_u32<<<(int)((n + 255) / 256), 256, 0, st>>>(p, n, v);
}

extern "C" void kernel_launch(void* const* d_in, const int* in_sizes, int n_in,
                              void* d_out, int out_size, void* d_ws, size_t ws_size,
                              hipStream_t stream) {
  auto F = [&](int i) -> const float* { return (const float*)d_in[i]; };

  const float* x_user = F(0);                 // NU x 64
  const float* x_prod = F(1);                 // NPR x 128
  const int*   eidx   = (const int*)d_in[2];  // 2 x E
  const int*   esrc   = eidx;
  const int*   edst   = eidx + EDG_;

  // params in _make_params() insertion order, flattened after the 3 tensors
  const float *user_norm_g = F(3),  *user_norm_b = F(4);
  const float *ul1_W = F(5),  *ul1_b = F(6);
  const float *ul2_W = F(7),  *ul2_b = F(8);
  const float *ul3_W = F(9),  *ul3_b = F(10);
  const float *ubn1_g = F(11), *ubn1_b = F(12);
  const float *ubn2_g = F(13), *ubn2_b = F(14);
  const float *prod_norm_g = F(15), *prod_norm_b = F(16);
  const float *pl1_W = F(17), *pl1_b = F(18);
  const float *pl2_W = F(19), *pl2_b = F(20);
  const float *pl3_W = F(21), *pl3_b = F(22);
  const float *pbn1_g = F(23), *pbn1_b = F(24);
  const float *pbn2_g = F(25), *pbn2_b = F(26);
  const float *attn_v_W = F(27), *attn_v_b = F(28);
  const float *attn_o_W = F(29), *attn_o_b = F(30);
  const float* sage_l_W[3] = { F(31), F(36), F(41) };
  const float* sage_l_b[3] = { F(32), F(37), F(42) };
  const float* sage_r_W[3] = { F(33), F(38), F(43) };
  const float* ln_g[3]     = { F(34), F(39), F(44) };
  const float* ln_b[3]     = { F(35), F(40), F(45) };
  const float* gat_l_W[3] = { F(46), nullptr, F(52) };
  const float* gat_l_b[3] = { F(47), nullptr, F(53) };
  const float* gat_r_W[3] = { F(48), nullptr, F(54) };
  const float* gat_r_b[3] = { F(49), nullptr, F(55) };
  const float* gat_att[3] = { F(50), nullptr, F(56) };
  const float* gat_bias[3]= { F(51), nullptr, F(57) };
  const float *tc_q_W = F(58), *tc_q_b = F(59);
  const float *tc_k_W = F(60), *tc_k_b = F(61);
  const float *tc_v_W = F(62), *tc_v_b = F(63);
  const float *tc_skip_W = F(64), *tc_skip_b = F(65);
  const float *fc1_W = F(66), *fc1_b = F(67);
  const float *fc2_W = F(68), *fc2_b = F(69);
  const float *fc3_W = F(70), *fc3_b = F(71);
  const float *fcbn1_g = F(72), *fcbn1_b = F(73);
  const float *fcbn2_g = F(74), *fcbn2_b = F(75);
  const float *outln_g = F(76), *outln_b = F(77);

  // ---------------- workspace layout (floats) ----------------
  const size_t F128N = (size_t)NN_ * HID_;   // 12.8M floats
  const size_t F128U = (size_t)NU_ * HID_;   // 6.4M floats
  float* ws  = (float*)d_ws;
  float* X   = ws;                 // node features [N x 128]
  float* XUK = X + F128N;          // user-tower output kept for head [NU x 128]
  float* S   = XUK + F128U;        // phase-reused scratch
  // towers (phase 1)
  float* uid = S;
  float* t2  = S + 1 * F128U;      // t2 -> xid in place
  float* t3  = S + 2 * F128U;      // t3 -> xu in place
  float* av  = S + 3 * F128U;
  float* t1  = S + 4 * F128U;      // NPR x 256
  float* pid = S + 6 * F128U;
  float* t3p = S + 7 * F128U;
  // message passing (phase 2)
  float* agg = S;                  // also attention output accumulator
  float* xsg = S + 1 * F128N;
  float* xlq = S + 2 * F128N;      // gat xl / tconv q
  float* xrk = S + 3 * F128N;      // gat xr / tconv k
  float* vbf = S + 4 * F128N;      // tconv v
  float* deg = S + 5 * F128N;                          // [N] (recip degree)
  unsigned* mU = (unsigned*)(deg + NN_);               // [N*4]
  float* zB  = deg + 5 * (size_t)NN_;                  // [N*4]
  float* esc = deg + 9 * (size_t)NN_;                  // [(E+N)*4]
  float* bnS = esc + 4 * (size_t)ELP_;                 // [256] folded BN scale
  float* bnB = bnS + 256;                              // [256] folded BN shift
  // head (phase 3)
  float* ue = S;                   // NU x 256
  float* h1 = S + 1 * F128N;       // NU x 256
  float* o1 = S + 2 * F128N;       // NU x 128

  const unsigned NEG_INF_ORD = 0x007FFFFFu;  // f2o(-inf)

  // ========================== user tower ==========================
  STATS(stream, x_user, NU_, 64, user_norm_g, user_norm_b, bnS, bnB);
  G<64, PRE_BN>(stream, x_user, NU_, ul1_W, ul1_b, uid, 128, bnS, bnB);   // uid
  G<128, PRE_GELU>(stream, uid, NU_, ul2_W, ul2_b, t2, 128);              // t2
  STATS(stream, t2, NU_, 128, ubn1_g, ubn1_b, bnS, bnB);
  bn_res_gelu_kernel<<<(int)((F128U + 255) / 256), 256, 0, stream>>>(
      t2, uid, bnS, bnB, t2, (long)F128U, 128);                           // xid
  G<128, PRE_NONE>(stream, t2, NU_, ul3_W, ul3_b, t3, 128);               // t3
  STATS(stream, t3, NU_, 128, ubn2_g, ubn2_b, bnS, bnB);
  bn_res_gelu_kernel<<<(int)((F128U + 255) / 256), 256, 0, stream>>>(
      t3, t2, bnS, bnB, t3, (long)F128U, 128);                            // xu
  G<128, PRE_NONE>(stream, t3, NU_, attn_v_W, attn_v_b, av, 128);
  G<128, PRE_NONE>(stream, av, NU_, attn_o_W, attn_o_b, XUK, 128,
                   nullptr, nullptr, nullptr, t3, 1.0f);                  // xu+attn
  hipMemcpyAsync(X, XUK, sizeof(float) * F128U, hipMemcpyDeviceToDevice, stream);

  // ========================= product tower ========================
  STATS(stream, x_prod, NPR_, 128, prod_norm_g, prod_norm_b, bnS, bnB);
  G<128, PRE_BN>(stream, x_prod, NPR_, pl1_W, pl1_b, t1, 256, bnS, bnB);  // t1
  STATS(stream, t1, NPR_, 256, pbn1_g, pbn1_b, bnS, bnB);
  G<256, PRE_BN_GELU>(stream, t1, NPR_, pl2_W, pl2_b, pid, 128, bnS, bnB);// pid
  G<128, PRE_GELU>(stream, pid, NPR_, pl3_W, pl3_b, t3p, 128);            // t3p
  STATS(stream, t3p, NPR_, 128, pbn2_g, pbn2_b, bnS, bnB);
  bn_res_gelu_kernel<<<(int)((F128U + 255) / 256), 256, 0, stream>>>(
      t3p, pid, bnS, bnB, X + F128U, (long)F128U, 128);                   // x[NU:]

  // ======================= degrees (once) =========================
  FILLF(stream, deg, NN_, 0.f);
  deg_kernel<<<(EDG_ + 255) / 256, 256, 0, stream>>>(edst, EDG_, deg);
  recip_deg_kernel<<<(NN_ + 255) / 256, 256, 0, stream>>>(deg, NN_);

  // ===================== message passing rounds ===================
  for (int i = 0; i < 3; ++i) {
    // ---- SAGE ----
    FILLF(stream, agg, (long)F128N, 0.f);
    scatter_agg_kernel<<<(int)(((long)EDG_ * 128 + 255) / 256), 256, 0, stream>>>(
        X, esrc, edst, agg, (long)EDG_ * 128);
    G<128, PRE_NONE>(stream, X, NN_, sage_r_W[i], nullptr, xsg, 128);
    G<128, PRE_ROWSCALE>(stream, agg, NN_, sage_l_W[i], sage_l_b[i], xsg, 128,
                         nullptr, nullptr, deg, xsg, 1.0f);
    // ---- attention branch ----
    if (i == 1) {  // TransformerConv
      G<128, PRE_NONE>(stream, X, NN_, tc_q_W, tc_q_b, xlq, 128);
      G<128, PRE_NONE>(stream, X, NN_, tc_k_W, tc_k_b, xrk, 128);
      G<128, PRE_NONE>(stream, X, NN_, tc_v_W, tc_v_b, vbf, 128);
      FILLU(stream, mU, (long)NN_ * 4, NEG_INF_ORD);
      FILLF(stream, zB, (long)NN_ * 4, 0.f);
      FILLF(stream, agg, (long)F128N, 0.f);
      tc_score_kernel<<<(EDG_ * 4 + 255) / 256, 256, 0, stream>>>(
          xlq, xrk, esrc, edst, EDG_, esc, mU);
      seg_exp_kernel<<<(EDG_ * 4 + 255) / 256, 256, 0, stream>>>(
          edst, EDG_, 0, mU, esc, zB);
      seg_agg_kernel<<<(int)(((long)EDG_ * 128 + 255) / 256), 256, 0, stream>>>(
          vbf, esrc, edst, EDG_, 0, esc, zB, agg);
      // fold skip projection into the attention output: agg += tc_skip(x)+b
      G<128, PRE_NONE>(stream, X, NN_, tc_skip_W, tc_skip_b, agg, 128,
                       nullptr, nullptr, nullptr, agg, 1.0f);
      combine_kernel<<<NN_, 128, 0, stream>>>(X, xsg, agg, nullptr, ln_g[i], ln_b[i]);
    } else {       // GATv2 (with self loops)
      G<128, PRE_NONE>(stream, X, NN_, gat_l_W[i], gat_l_b[i], xlq, 128);
      G<128, PRE_NONE>(stream, X, NN_, gat_r_W[i], gat_r_b[i], xrk, 128);
      FILLU(stream, mU, (long)NN_ * 4, NEG_INF_ORD);
      FILLF(stream, zB, (long)NN_ * 4, 0.f);
      FILLF(stream, agg, (long)F128N, 0.f);
      gat_score_kernel<<<(ELP_ * 4 + 255) / 256, 256, 0, stream>>>(
          xlq, xrk, gat_att[i], esrc, edst, EDG_, NN_, esc, mU);
      seg_exp_kernel<<<(ELP_ * 4 + 255) / 256, 256, 0, stream>>>(
          edst, EDG_, NN_, mU, esc, zB);
      seg_agg_kernel<<<(int)(((long)ELP_ * 128 + 255) / 256), 256, 0, stream>>>(
          xlq, esrc, edst, EDG_, NN_, esc, zB, agg);
      combine_kernel<<<NN_, 128, 0, stream>>>(X, xsg, agg, gat_bias[i], ln_g[i], ln_b[i]);
    }
  }

  // ============================ head ==============================
  concat_kernel<<<(int)(((long)NU_ * 256 + 255) / 256), 256, 0, stream>>>(X, XUK, ue, NU_);
  G<256, PRE_NONE>(stream, ue, NU_, fc1_W, fc1_b, h1, 256);
  STATS(stream, h1, NU_, 256, fcbn1_g, fcbn1_b, bnS, bnB);
  G<256, PRE_BN_GELU>(stream, h1, NU_, fc2_W, fc2_b, o1, 128, bnS, bnB);
  STATS(stream, o1, NU_, 128, fcbn2_g, fcbn2_b, bnS, bnB);
  head_kernel<<<(NU_ + 127) / 128, 128, 0, stream>>>(
      o1, bnS, bnB, fc3_W, fc3_b, outln_g, outln_b, (float*)d_out, NU_);

  (void)in_sizes; (void)n_in; (void)out_size; (void)ws_size;
}